// PagedAttention_39797166965454
// MI455X (gfx1250) — compile-verified
//
#include <hip/hip_runtime.h>

typedef float v2f __attribute__((ext_vector_type(2)));
typedef float v8f __attribute__((ext_vector_type(8)));

namespace {
constexpr int   kT     = 64;
constexpr int   kH     = 32;
constexpr int   kD     = 128;
constexpr int   kKVH   = 8;
constexpr int   kG     = 4;     // query heads per kv head
constexpr int   kBS    = 16;    // page (block) size
constexpr int   kMaxB  = 128;   // max blocks per seq
constexpr int   kWaves = 8;     // waves per workgroup (flash-decode split)
constexpr float kScale = 0.08838834764831845f;
constexpr float kNegInf = -1e30f;
}

static __device__ __forceinline__ float sx16(float v) {
  return __shfl_xor(v, 16, 32);
}

// Process one 16-token KV page with fp32 WMMA + online softmax.
// LAST=true only for the single page holding the current token (wave-uniform),
// so the hot path (LAST=false) has no masking and no pointer selects.
template <bool LAST>
static __device__ __forceinline__ void process_page(
    const float* __restrict__ kpage, const float* __restrict__ vpage,
    const float* __restrict__ kfresh, const float* __restrict__ vfresh,
    const v2f* __restrict__ qb,
    int ln, int hi, int page, int ctx, int o_last,
    float& m, float& l, v8f (&O)[8]) {
  // ---- scores C[t][g] = sum_d K[t][d] * Q[g][d], 32 fp32 WMMAs (K dim = 4) ----
  v8f cacc = {0.f, 0.f, 0.f, 0.f, 0.f, 0.f, 0.f, 0.f};
  #pragma unroll
  for (int kc = 0; kc < 32; ++kc) {
    // A[t][k]: lane t = ln loads pair K[t][4kc+2hi .. +1] (contiguous in x-dim)
    const float* ka = kpage + (kc >> 1) * 128 + (kc & 1) * 4 + ln * 8 + 2 * hi;
    v2f av;
    if (LAST) {
      if (ln == o_last) ka = kfresh + 4 * kc + 2 * hi;  // fresh current-token K
      av = *(const v2f*)ka;
    } else {
      av = __builtin_nontemporal_load((const v2f*)ka);  // stream-once data
    }
    cacc = __builtin_amdgcn_wmma_f32_16x16x4_f32(
        false, av, false, qb[kc], (short)0, cacc, false, false);
  }

  // ---- online softmax (head = ln across lanes; tokens r + 8*hi in vgpr idx) ----
  float pv[8];
  float bm = kNegInf;
  #pragma unroll
  for (int r = 0; r < 8; ++r) {
    float s = cacc[r] * kScale;
    if (LAST) {
      const int t = page * kBS + r + hi * 8;
      if (t >= ctx) s = kNegInf;
    }
    pv[r] = s;
    bm = fmaxf(bm, s);
  }
  bm = fmaxf(bm, sx16(bm));               // combine both token halves
  const float m_new = fmaxf(m, bm);
  const float alpha = __expf(m - m_new);
  float bl = 0.f;
  #pragma unroll
  for (int r = 0; r < 8; ++r) {
    float e = __expf(pv[r] - m_new);
    if (LAST) {
      const int t = page * kBS + r + hi * 8;
      if (t >= ctx) e = 0.f;
    }
    pv[r] = e;
    bl += e;
  }
  bl += sx16(bl);
  l = l * alpha + bl;
  m = m_new;
  #pragma unroll
  for (int c = 0; c < 8; ++c) O[c] = O[c] * alpha;  // rescale accumulator

  // ---- convert P (WMMA C-layout) into B fragments for the 4 token groups ----
  const float s0 = sx16(pv[0]), s1 = sx16(pv[1]), s2 = sx16(pv[2]), s3 = sx16(pv[3]);
  const float s4 = sx16(pv[4]), s5 = sx16(pv[5]), s6 = sx16(pv[6]), s7 = sx16(pv[7]);
  v2f pB[4];
  pB[0] = hi ? (v2f){s2, s3}       : (v2f){pv[0], pv[1]};   // tokens 0..3
  pB[1] = hi ? (v2f){s6, s7}       : (v2f){pv[4], pv[5]};   // tokens 4..7
  pB[2] = hi ? (v2f){pv[2], pv[3]} : (v2f){s0, s1};         // tokens 8..11
  pB[3] = hi ? (v2f){pv[6], pv[7]} : (v2f){s4, s5};         // tokens 12..15

  // ---- O[d][g] += V^T (A) x P (B): 8 d-chunks x 4 token groups ----
  #pragma unroll
  for (int c = 0; c < 8; ++c) {
    const int d = 16 * c + ln;
    const float* vbase = vpage + (size_t)d * kBS;   // V[t][d] at vpage[d*16 + t]
    float vf = 0.f;
    if (LAST) vf = vfresh[d];                       // fresh current-token V
    #pragma unroll
    for (int j = 0; j < 4; ++j) {
      const int t0 = 4 * j + 2 * hi;
      v2f av;
      if (LAST) {
        av = *(const v2f*)(vbase + t0);             // {V[t0][d], V[t0+1][d]}
        if (t0 == o_last)     av.x = vf;
        if (t0 + 1 == o_last) av.y = vf;
      } else {
        av = __builtin_nontemporal_load((const v2f*)(vbase + t0));
      }
      O[c] = __builtin_amdgcn_wmma_f32_16x16x4_f32(
          false, av, false, pB[j], (short)0, O[c], false, false);
    }
  }
}

__global__ __launch_bounds__(kWaves * 32) void paged_attn_wmma(
    const float* __restrict__ query,        // [T, H*D]
    const float* __restrict__ key,          // [T, KVH*D]
    const float* __restrict__ value,        // [T, KVH*D]
    const float* __restrict__ key_cache,    // [NB, KVH, D/8, BS, 8]
    const float* __restrict__ value_cache,  // [NB, KVH, D, BS]
    const int*  __restrict__ block_tables,  // [T, kMaxB]
    const int*  __restrict__ context_lens,  // [T]
    float* __restrict__ out)                // [T, H*D]
{
  __shared__ float sm_acc[8 * 8 * 32];      // merged O: [d-chunk][row][lane]
  __shared__ float sm_m[kWaves * 32];
  __shared__ float sm_l[kWaves * 32];

  const int kvh  = blockIdx.x;
  const int seq  = blockIdx.y;
  const int tid  = threadIdx.x;
  const int wave = tid >> 5;
  const int lane = tid & 31;
  const int ln   = lane & 15;   // head col (B/C N-index) OR token/dim row (A M-index)
  const int hi   = lane >> 4;   // upper half of wave

  // zero the shared merge accumulator (covered by the single pre-merge barrier)
  #pragma unroll
  for (int i = 0; i < 8; ++i) sm_acc[tid + i * 256] = 0.f;

  const int ctx       = context_lens[seq];
  const int num_pages = (ctx + kBS - 1) / kBS;
  const int last      = ctx - 1;          // slot of the current token
  const int p_last    = last / kBS;       // == num_pages - 1
  const int o_last    = last % kBS;

  // ---- resident Q fragments: B operand of QK wmma ----
  // B[k][g] for head-dim chunk kc: vgpr0 = {Q[g][4kc] | Q[g][4kc+2]}, vgpr1 = {+1 | +3}
  v2f qb[32];
  {
    const float* qrow =
        query + (size_t)seq * (kH * kD) + (size_t)(kvh * kG + ln) * kD + 2 * hi;
    #pragma unroll
    for (int kc = 0; kc < 32; ++kc) {
      v2f qv = {0.f, 0.f};
      if (ln < kG) qv = *(const v2f*)(qrow + 4 * kc);   // zero rows for g>=4
      qb[kc] = qv;
    }
  }

  // fresh (current-token) K/V rows for this (seq, kv_head)
  const float* kfresh = key   + ((size_t)seq * kKVH + kvh) * kD;
  const float* vfresh = value + ((size_t)seq * kKVH + kvh) * kD;

  float m = kNegInf;
  float l = 0.f;
  v8f O[8];   // O[d-chunk]: C-layout, d = 16c + r + 8*hi, head = ln
  #pragma unroll
  for (int c = 0; c < 8; ++c) O[c] = {0.f, 0.f, 0.f, 0.f, 0.f, 0.f, 0.f, 0.f};

  for (int page = wave; page < num_pages; page += kWaves) {
    const int blk = block_tables[seq * kMaxB + page];
    const float* kpage = key_cache   + ((size_t)blk * kKVH + kvh) * (kD * kBS);
    const float* vpage = value_cache + ((size_t)blk * kKVH + kvh) * (kD * kBS);
    if (page == p_last) {   // wave-uniform branch; cold path with fixups
      process_page<true >(kpage, vpage, kfresh, vfresh, qb,
                          ln, hi, page, ctx, o_last, m, l, O);
    } else {                // hot path: no masking, no selects, NT streaming loads
      process_page<false>(kpage, vpage, kfresh, vfresh, qb,
                          ln, hi, page, ctx, o_last, m, l, O);
    }
  }

  // ---- cross-wave flash-decode merge via LDS ----
  sm_m[wave * 32 + lane] = m;
  sm_l[wave * 32 + lane] = l;
  __syncthreads();

  float mstar = kNegInf;
  #pragma unroll
  for (int w = 0; w < kWaves; ++w) mstar = fmaxf(mstar, sm_m[w * 32 + lane]);
  float L = 0.f;
  #pragma unroll
  for (int w = 0; w < kWaves; ++w) {
    const float lw = sm_l[w * 32 + lane];
    const float sw = (lw > 0.f) ? __expf(sm_m[w * 32 + lane] - mstar) : 0.f;
    L += sw * lw;
  }
  const float sc_self = (l > 0.f) ? __expf(m - mstar) : 0.f;

  #pragma unroll
  for (int c = 0; c < 8; ++c)
    #pragma unroll
    for (int r = 0; r < 8; ++r)
      atomicAdd(&sm_acc[c * 256 + r * 32 + lane], O[c][r] * sc_self);  // ds_add_f32
  __syncthreads();

  // wave w writes d-chunk c = w; valid head lanes only (g = ln < 4)
  if (ln < kG) {
    const float inv = 1.f / L;
    const int c = wave;
    float* orow = out + (size_t)seq * (kH * kD) + (size_t)(kvh * kG + ln) * kD;
    #pragma unroll
    for (int r = 0; r < 8; ++r) {
      const int d = c * 16 + r + hi * 8;
      orow[d] = sm_acc[c * 256 + r * 32 + lane] * inv;
    }
  }
}

extern "C" void kernel_launch(void* const* d_in, const int* in_sizes, int n_in,
                              void* d_out, int out_size, void* d_ws, size_t ws_size,
                              hipStream_t stream) {
  const float* query        = (const float*)d_in[0];
  const float* key          = (const float*)d_in[1];
  const float* value        = (const float*)d_in[2];
  const float* key_cache    = (const float*)d_in[3];
  const float* value_cache  = (const float*)d_in[4];
  const int*   block_tables = (const int*)d_in[5];
  const int*   context_lens = (const int*)d_in[6];
  // d_in[7] = slot_mapping (derivable from context_lens + block_tables; unused)
  float* out = (float*)d_out;

  dim3 grid(kKVH, kT, 1);
  dim3 block(kWaves * 32, 1, 1);
  hipLaunchKernelGGL(paged_attn_wmma, grid, block, 0, stream,
                     query, key, value, key_cache, value_cache,
                     block_tables, context_lens, out);
}